// LinearAttention_71287867179560
// MI455X (gfx1250) — compile-verified
//
#include <hip/hip_runtime.h>
#include <hip/hip_bf16.h>

#if defined(__HIP_DEVICE_COMPILE__) && !__has_builtin(__builtin_amdgcn_wmma_f32_16x16x4_f32)
#error "missing __builtin_amdgcn_wmma_f32_16x16x4_f32 (device pass)"
#endif

typedef float v2f __attribute__((ext_vector_type(2)));
typedef float v8f __attribute__((ext_vector_type(8)));

#if defined(__HIP_DEVICE_COMPILE__)
#define WMMA_F32(a, b, c) \
    __builtin_amdgcn_wmma_f32_16x16x4_f32(false, (a), false, (b), (short)0, (c), false, false)
#else
// Host pass never executes device code; stub keeps the host compiler happy.
#define WMMA_F32(a, b, c) (c)
#endif

// 16x4 f32 A-fragment (also used as B when the NT operand is contiguous along K):
// lanes 0-15: row = row0+lane,    k = {k0,  k0+1}
// lanes 16-31: row = row0+lane-16, k = {k0+2, k0+3}
__device__ __forceinline__ v2f frag_contig(const float* __restrict__ base,
                                           int row0, int k0, int ld, int lane) {
    int r  = row0 + (lane & 15);
    int kb = k0 + ((lane >> 4) << 1);
    return *(const v2f*)(base + (size_t)r * ld + kb);
}

// Fragment where K is the slow (strided) dimension: element [k][col] = base[k*ld + col]
__device__ __forceinline__ v2f frag_strided(const float* __restrict__ base,
                                            int k0, int col0, int ld, int lane) {
    int c  = col0 + (lane & 15);
    int kb = k0 + ((lane >> 4) << 1);
    const float* p = base + (size_t)kb * ld + c;
    v2f v;
    v.x = p[0];
    v.y = p[ld];
    return v;
}

// out[m][n] = act( sum_k A[m][k] * W[n][k] + bias[n] )   (torch Linear: A @ W^T + b)
// One wave per 64x64 output tile, 4x4 grid of 16x16 f32 WMMA accumulators.
__global__ __launch_bounds__(32) void gemm_nt_bias_act_64x64(
    const float* __restrict__ A, const float* __restrict__ W,
    const float* __restrict__ bias, float* __restrict__ out,
    int M, int Nout, int K, int act)
{
    int lane   = threadIdx.x;
    int ntiles = Nout >> 6;
    int tm = (blockIdx.x / ntiles) << 6;
    int tn = (blockIdx.x % ntiles) << 6;

    v8f zero = {};
    v8f acc[4][4];
#pragma unroll
    for (int i = 0; i < 4; ++i)
#pragma unroll
        for (int j = 0; j < 4; ++j) acc[i][j] = zero;

    for (int k0 = 0; k0 < K; k0 += 4) {
        v2f af[4], bf[4];
#pragma unroll
        for (int i = 0; i < 4; ++i) af[i] = frag_contig(A, tm + i * 16, k0, K, lane);
#pragma unroll
        for (int j = 0; j < 4; ++j) bf[j] = frag_contig(W, tn + j * 16, k0, K, lane);
#pragma unroll
        for (int i = 0; i < 4; ++i)
#pragma unroll
            for (int j = 0; j < 4; ++j)
                acc[i][j] = WMMA_F32(af[i], bf[j], acc[i][j]);
    }

    // C/D layout: VGPR r -> (M = r + 8*(lane>>4), N = lane&15)
    int cb  = lane & 15;
    int rhi = (lane >> 4) << 3;
#pragma unroll
    for (int j = 0; j < 4; ++j) {
        int col  = tn + j * 16 + cb;
        float bv = bias[col];
#pragma unroll
        for (int i = 0; i < 4; ++i) {
#pragma unroll
            for (int r = 0; r < 8; ++r) {
                int row = tm + i * 16 + rhi + r;
                float v = acc[i][j][r] + bv;
                if (act) v = (v > 0.0f) ? (v + 1.0f) : __expf(v);  // elu(v)+1
                out[(size_t)row * Nout + col] = v;
            }
        }
    }
}

// Chunked causal linear attention. One block per (b,h); 4 waves; chunk = 64 = d.
// Per chunk:  P = causal_mask(Qc Kc^T);  O = Qc @ S + P @ Vc;  S += Kc^T @ Vc.
__global__ __launch_bounds__(128) void linattn_chunked(
    const float* __restrict__ Q, const float* __restrict__ K,
    const float* __restrict__ V, float* __restrict__ O,
    int N, int E, int H)
{
    const int d  = E / H;          // 64
    int bh = blockIdx.x;
    int b  = bh / H, h = bh % H;
    const float* Qh = Q + (size_t)b * N * E + h * d;
    const float* Kh = K + (size_t)b * N * E + h * d;
    const float* Vh = V + (size_t)b * N * E + h * d;
    float*       Oh = O + (size_t)b * N * E + h * d;

    __shared__ float S[64][64];     // running k v^T state (d x d)
    __shared__ float P[4][16][66];  // per-wave masked QK^T strip, padded vs bank conflicts

    int lane = threadIdx.x & 31;
    int w    = threadIdx.x >> 5;    // wave id 0..3
    int m0   = w << 4;              // this wave's 16-row strip

    for (int i = threadIdx.x; i < 64 * 64; i += 128) (&S[0][0])[i] = 0.0f;
    __syncthreads();

    v8f zero = {};
    int cb  = lane & 15;
    int rhi = (lane >> 4) << 3;

    for (int c0 = 0; c0 < N; c0 += 64) {
        const float* Qc = Qh + (size_t)c0 * E;
        const float* Kc = Kh + (size_t)c0 * E;
        const float* Vc = Vh + (size_t)c0 * E;

        // ---- P strip = rows [m0,m0+16) of Qc @ Kc^T, causal-masked, staged to LDS
        v8f p[4];
#pragma unroll
        for (int j = 0; j < 4; ++j) p[j] = zero;
        for (int k0 = 0; k0 < 64; k0 += 4) {
            v2f aq = frag_contig(Qc, m0, k0, E, lane);
#pragma unroll
            for (int j = 0; j < 4; ++j) {
                v2f bk = frag_contig(Kc, j * 16, k0, E, lane);  // NT: B = K rows
                p[j] = WMMA_F32(aq, bk, p[j]);
            }
        }
#pragma unroll
        for (int j = 0; j < 4; ++j) {
            int jc = j * 16 + cb;
#pragma unroll
            for (int r = 0; r < 8; ++r) {
                int i = rhi + r;  // row within strip
                P[w][i][jc] = (jc <= m0 + i) ? p[j][r] : 0.0f;
            }
        }

        // ---- O strip = Qc_strip @ S  +  P_strip @ Vc
        v8f o[4];
#pragma unroll
        for (int j = 0; j < 4; ++j) o[j] = zero;
        for (int k0 = 0; k0 < 64; k0 += 4) {
            v2f aq = frag_contig(Qc, m0, k0, E, lane);
#pragma unroll
            for (int j = 0; j < 4; ++j) {
                v2f bs = frag_strided(&S[0][0], k0, j * 16, 64, lane);
                o[j] = WMMA_F32(aq, bs, o[j]);
            }
        }
        for (int k0 = 0; k0 < 64; k0 += 4) {
            v2f ap = frag_contig(&P[w][0][0], 0, k0, 66, lane);
#pragma unroll
            for (int j = 0; j < 4; ++j) {
                v2f bv = frag_strided(Vc, k0, j * 16, E, lane);
                o[j] = WMMA_F32(ap, bv, o[j]);
            }
        }
#pragma unroll
        for (int j = 0; j < 4; ++j)
#pragma unroll
            for (int r = 0; r < 8; ++r)
                Oh[(size_t)(c0 + m0 + rhi + r) * E + j * 16 + cb] = o[j][r];

        __syncthreads();  // all waves finished reading S for this chunk

        // ---- S rows [m0,m0+16) += Kc^T_strip @ Vc
        v8f dS[4];
#pragma unroll
        for (int j = 0; j < 4; ++j) dS[j] = zero;
        for (int k0 = 0; k0 < 64; k0 += 4) {                       // k0 runs over chunk time
            v2f ak = frag_strided(Kc, k0, m0, E, lane);            // A[k1][t] = K[t][k1]
#pragma unroll
            for (int j = 0; j < 4; ++j) {
                v2f bv = frag_strided(Vc, k0, j * 16, E, lane);
                dS[j] = WMMA_F32(ak, bv, dS[j]);
            }
        }
#pragma unroll
        for (int j = 0; j < 4; ++j)
#pragma unroll
            for (int r = 0; r < 8; ++r)
                S[m0 + rhi + r][j * 16 + cb] += dS[j][r];

        __syncthreads();  // state visible before next chunk
    }
}

extern "C" void kernel_launch(void* const* d_in, const int* in_sizes, int n_in,
                              void* d_out, int out_size, void* d_ws, size_t ws_size,
                              hipStream_t stream)
{
    (void)in_sizes; (void)n_in; (void)out_size; (void)ws_size;

    const int Bb = 2, N = 2048, E = 1024, H = 16;
    const int M = Bb * N;  // 4096 rows

    const float* x  = (const float*)d_in[0];
    const float* Wq = (const float*)d_in[1]; const float* bq = (const float*)d_in[2];
    const float* Wk = (const float*)d_in[3]; const float* bk = (const float*)d_in[4];
    const float* Wv = (const float*)d_in[5]; const float* bv = (const float*)d_in[6];
    const float* Wo = (const float*)d_in[7]; const float* bo = (const float*)d_in[8];
    float* out = (float*)d_out;

    size_t tensor = (size_t)M * E;          // 4M floats each
    float* Qb = (float*)d_ws;               // ws: Q | K | V | attn-out = 64 MB
    float* Kb = Qb + tensor;
    float* Vb = Kb + tensor;
    float* Ab = Vb + tensor;

    dim3 gproj((M / 64) * (E / 64));        // 1024 wave-tiles
    gemm_nt_bias_act_64x64<<<gproj, 32, 0, stream>>>(x,  Wq, bq, Qb, M, E, E, 1);
    gemm_nt_bias_act_64x64<<<gproj, 32, 0, stream>>>(x,  Wk, bk, Kb, M, E, E, 1);
    gemm_nt_bias_act_64x64<<<gproj, 32, 0, stream>>>(x,  Wv, bv, Vb, M, E, E, 0);

    linattn_chunked<<<dim3(Bb * H), 128, 0, stream>>>(Qb, Kb, Vb, Ab, N, E, H);

    gemm_nt_bias_act_64x64<<<gproj, 32, 0, stream>>>(Ab, Wo, bo, out, M, E, E, 0);
}